// CustomGATConv_58437325029515
// MI455X (gfx1250) — compile-verified
//
#include <hip/hip_runtime.h>
#include <hip/hip_bf16.h>

// ---------------------------------------------------------------------------
// GATConv on MI455X (gfx1250, wave32).
//   h = x@W via V_WMMA_F32_16X16X4_F32 (fp32-exact matrix core path)
//   edge softmax + scatter via L2-resident float atomics (all state << 192MB L2)
// ---------------------------------------------------------------------------

#define D 128           // D_IN == D_OUT == 128
#define NEG_SLOPE 0.2f

typedef float v2f __attribute__((ext_vector_type(2)));
typedef float v8f __attribute__((ext_vector_type(8)));

// Monotonic float <-> signed-int bijection (for exact atomicMax on floats).
__device__ __forceinline__ int f2ord(float f) {
    int i = __float_as_int(f);
    return (i < 0) ? (i ^ 0x7FFFFFFF) : i;
}
__device__ __forceinline__ float ord2f(int i) {
    return __int_as_float((i < 0) ? (i ^ 0x7FFFFFFF) : i);
}

// ---------------------------------------------------------------------------
// out[n*128+d] = bias[d]; maxb = ord(-inf); sumb = 0
// ---------------------------------------------------------------------------
__global__ void gat_init(float* __restrict__ out, const float* __restrict__ bias,
                         int* __restrict__ maxb, float* __restrict__ sumb, int N) {
    int i = blockIdx.x * blockDim.x + threadIdx.x;
    if (i < N * D) out[i] = bias[i & (D - 1)];
    if (i < N) { maxb[i] = (int)0x807FFFFF; /* ord(-inf) */ sumb[i] = 0.0f; }
}

// ---------------------------------------------------------------------------
// h = x @ W.  One block = 16-row strip; wave w handles cols [16w, 16w+16).
// A frag (16x4 f32): lane l -> row l%16, VGPR0=K+2*(l/16), VGPR1=K+2*(l/16)+1
// B frag (4x16 f32): lane l -> col l%16, same K split.
// C/D   (16x16 f32): VGPR j -> row j + 8*(l/16), col l%16.
// ---------------------------------------------------------------------------
__global__ __launch_bounds__(256)
void gat_gemm_wmma(const float* __restrict__ x, const float* __restrict__ W,
                   float* __restrict__ h, int N) {
    const int lane = threadIdx.x & 31;
    const int wave = threadIdx.x >> 5;
    const int m0   = blockIdx.x * 16;
    const int n0   = wave * 16;
    const int mr   = lane & 15;          // row (A) / col (B) within tile
    const int kh   = (lane >> 4) << 1;   // 0 or 2: K sub-phase per half-wave

    const int arow = min(m0 + mr, N - 1);          // clamp (N is 16-aligned here)
    const float* __restrict__ xr = x + (size_t)arow * D;

    v8f c = {};
    #pragma unroll
    for (int k = 0; k < D; k += 4) {
        v2f a, b;
        a.x = xr[k + kh];
        a.y = xr[k + kh + 1];
        b.x = W[(size_t)(k + kh) * D + n0 + mr];
        b.y = W[(size_t)(k + kh + 1) * D + n0 + mr];
        c = __builtin_amdgcn_wmma_f32_16x16x4_f32(
                /*neg_a=*/false, a, /*neg_b=*/false, b,
                /*c_mod=*/(short)0, c, /*reuse_a=*/false, /*reuse_b=*/false);
    }

    const int col   = n0 + mr;
    const int rbase = m0 + ((lane >> 4) << 3);
    #pragma unroll
    for (int j = 0; j < 8; ++j) {
        int row = rbase + j;
        if (row < N) h[(size_t)row * D + col] = c[j];
    }
}

// ---------------------------------------------------------------------------
// a_s[i] = h[i]·att_src, a_d[i] = h[i]·att_dst. One wave per node.
// ---------------------------------------------------------------------------
__global__ __launch_bounds__(256)
void gat_logits(const float* __restrict__ h, const float* __restrict__ att_s,
                const float* __restrict__ att_d, float* __restrict__ as_,
                float* __restrict__ ad_, int N) {
    int node = blockIdx.x * 8 + (threadIdx.x >> 5);
    if (node >= N) return;
    int lane = threadIdx.x & 31;
    float4 v = ((const float4*)(h + (size_t)node * D))[lane];
    float4 s = ((const float4*)att_s)[lane];
    float4 d = ((const float4*)att_d)[lane];
    float ps = v.x * s.x + v.y * s.y + v.z * s.z + v.w * s.w;
    float pd = v.x * d.x + v.y * d.y + v.z * d.z + v.w * d.w;
    #pragma unroll
    for (int off = 16; off; off >>= 1) {
        ps += __shfl_xor(ps, off, 32);
        pd += __shfl_xor(pd, off, 32);
    }
    if (lane == 0) { as_[node] = ps; ad_[node] = pd; }
}

// ---------------------------------------------------------------------------
// Pass 1: e = leakyrelu(a_s[src]+a_d[dst]); stash e; segment-max into maxb.
// ---------------------------------------------------------------------------
__global__ __launch_bounds__(256)
void gat_edge_max(const int* __restrict__ ei, const float* __restrict__ as_,
                  const float* __restrict__ ad_, float* __restrict__ ebuf,
                  int* __restrict__ maxb, int E, int ET) {
    int e = blockIdx.x * blockDim.x + threadIdx.x;
    if (e >= ET) return;
    int s, d;
    if (e < E) { s = ei[e]; d = ei[E + e]; } else { s = d = e - E; }
    float t  = as_[s] + ad_[d];
    float le = (t > 0.0f) ? t : NEG_SLOPE * t;
    ebuf[e] = le;
    atomicMax(&maxb[d], f2ord(le));
}

// ---------------------------------------------------------------------------
// Pass 2: ex = exp(e - max[dst]); stash ex; segment-sum into sumb.
// ---------------------------------------------------------------------------
__global__ __launch_bounds__(256)
void gat_edge_exp(const int* __restrict__ ei, const int* __restrict__ maxb,
                  float* __restrict__ ebuf, float* __restrict__ sumb,
                  int E, int ET) {
    int e = blockIdx.x * blockDim.x + threadIdx.x;
    if (e >= ET) return;
    int d = (e < E) ? ei[E + e] : (e - E);
    float ex = __expf(ebuf[e] - ord2f(maxb[d]));
    ebuf[e] = ex;
    unsafeAtomicAdd(&sumb[d], ex);     // native global_atomic_add_f32
}

// ---------------------------------------------------------------------------
// Pass 3: out[dst] += (ex/sum[dst]) * h[src]. One wave per edge, float4/lane.
// All of h/out live in L2 (10 MB << 192 MB), so gathers+atomics are L2-speed.
// ---------------------------------------------------------------------------
__global__ __launch_bounds__(256)
void gat_aggregate(const int* __restrict__ ei, const float* __restrict__ h,
                   const float* __restrict__ ebuf, const float* __restrict__ sumb,
                   float* __restrict__ out, int E, int ET) {
    long long gt = (long long)blockIdx.x * blockDim.x + threadIdx.x;
    int e = (int)(gt >> 5);
    if (e >= ET) return;
    int lane = (int)(gt & 31);
    int s, d;
    if (e < E) { s = ei[e]; d = ei[E + e]; } else { s = d = e - E; }
    float alpha = ebuf[e] / sumb[d];
    float4 v = ((const float4*)(h + (size_t)s * D))[lane];
    float* ob = out + (size_t)d * D + lane * 4;
    unsafeAtomicAdd(ob + 0, alpha * v.x);
    unsafeAtomicAdd(ob + 1, alpha * v.y);
    unsafeAtomicAdd(ob + 2, alpha * v.z);
    unsafeAtomicAdd(ob + 3, alpha * v.w);
}

// ---------------------------------------------------------------------------
extern "C" void kernel_launch(void* const* d_in, const int* in_sizes, int n_in,
                              void* d_out, int out_size, void* d_ws, size_t ws_size,
                              hipStream_t stream) {
    const float* x     = (const float*)d_in[0];
    const float* W     = (const float*)d_in[1];
    const float* att_s = (const float*)d_in[2];
    const float* att_d = (const float*)d_in[3];
    const float* bias  = (const float*)d_in[4];
    const int*   ei    = (const int*)d_in[5];

    const int N  = in_sizes[0] / D;     // 10000
    const int E  = in_sizes[5] / 2;     // 640000
    const int ET = E + N;               // edges + self-loops

    // Workspace carve-up (all 16B-aligned: N*4 = 40000 bytes, multiple of 16).
    float* h    = (float*)d_ws;                 // N*D floats
    float* as_  = h + (size_t)N * D;            // N
    float* ad_  = as_ + N;                      // N
    int*   maxb = (int*)(ad_ + N);              // N
    float* sumb = (float*)(maxb + N);           // N
    float* ebuf = sumb + N;                     // ET
    float* out  = (float*)d_out;

    const int B = 256;
    gat_init<<<(N * D + B - 1) / B, B, 0, stream>>>(out, bias, maxb, sumb, N);
    gat_gemm_wmma<<<(N + 15) / 16, B, 0, stream>>>(x, W, h, N);
    gat_logits<<<(N + 7) / 8, B, 0, stream>>>(h, att_s, att_d, as_, ad_, N);
    gat_edge_max<<<(ET + B - 1) / B, B, 0, stream>>>(ei, as_, ad_, ebuf, maxb, E, ET);
    gat_edge_exp<<<(ET + B - 1) / B, B, 0, stream>>>(ei, maxb, ebuf, sumb, E, ET);
    long long aggT = (long long)ET * 32;
    gat_aggregate<<<(int)((aggT + B - 1) / B), B, 0, stream>>>(ei, h, ebuf, sumb, out, E, ET);
}